// TokenEmbedding_26886495273549
// MI455X (gfx1250) — compile-verified
//
#include <hip/hip_runtime.h>

typedef __attribute__((ext_vector_type(2))) float v2f;
typedef __attribute__((ext_vector_type(4))) float v4f;
typedef __attribute__((ext_vector_type(8))) float v8f;

#define SEQ    4096
#define CIN    7
#define NROWS  4096        // 7*585 + 1
#define HP     12289
#define GLEN   (SEQ * 3)   // 12288
#define KSTEPS 14          // K = 56 f32, 4 per wmma

// g[c, t] = x[min(t/3 + t%3, SEQ-1), c]  with zero outside [0, GLEN)
__device__ __forceinline__ float load_g(const float* __restrict__ x, int c, int t) {
  int s_  = t / 3;
  int j   = t - 3 * s_;
  int row = s_ + j;
  row = row < 0 ? 0 : row;
  row = row > (SEQ - 1) ? (SEQ - 1) : row;
  float v = x[row * CIN + c];
  return ((unsigned)t < (unsigned)GLEN) ? v : 0.0f;
}

__global__ __launch_bounds__(256) void conv_wmma_f32(
    const float* __restrict__ x,     // (4096, 7)
    const float* __restrict__ ker,   // (586, 8, 3)
    float* __restrict__ out)         // flat[h*4096 + r]
{
  const int lane  = threadIdx.x & 31;
  const int wave  = threadIdx.x >> 5;
  const int lhalf = lane & 15;
  const int hi16  = lane >> 4;     // 0: lanes 0-15, 1: lanes 16-31
  const int off   = hi16 * 2;      // K offset inside a 4-group (A/B layouts)

  const int h0  = blockIdx.x * 16;
  const int h   = h0 + lhalf;
  const bool hok = (h < HP);

  // ---- Build B tile (56 x 16 f32) once per wave; reused for 8 row-tiles ----
  // B layout per step s: V0 = {K=4s (lanes 0-15), K=4s+2 (lanes 16-31)},
  //                      V1 = {K=4s+1,            K=4s+3}
  v2f B[KSTEPS];
#pragma unroll
  for (int s = 0; s < KSTEPS; ++s) {
    int k0 = 4 * s + off;
    int k1 = k0 + 1;
    B[s].x = load_g(x, k0 >> 3, h - 4 + (k0 & 7));
    B[s].y = load_g(x, k1 >> 3, h - 4 + (k1 & 7));
  }

  const int tbase = blockIdx.y * 64 + wave * 8;   // 8 row-tiles per wave

  for (int i = 0; i < 8; ++i) {
    const int r0 = (tbase + i) * 16;
    const int r  = r0 + lhalf;                    // this lane's A-matrix row

    // channel / kernel id for this row (last row is kernel 585 on channel 0)
    int n = (r == NROWS - 1) ? 0 : (r / 585);
    int o = (r == NROWS - 1) ? 585 : (r - n * 585);

    // kernels[o, kh, 1] : the only surviving column of the 8x3 filter
    const float* kb = ker + o * 24 + 1;
    v2f A0, A1;                                   // steps 2n and 2n+1
    A0.x = kb[(off + 0) * 3];
    A0.y = kb[(off + 1) * 3];
    A1.x = kb[(off + 4) * 3];
    A1.y = kb[(off + 5) * 3];

    // Uniform range of populated K-steps in this tile (tile spans <= 2 channels)
    int nfirst = r0 / 585;
    int rlast  = r0 + 15;
    int nlast  = (rlast == NROWS - 1) ? 0 : (rlast / 585);
    int nlo = nfirst < nlast ? nfirst : nlast;
    int nhi = nfirst > nlast ? nfirst : nlast;
    int slo = 2 * nlo, shi = 2 * nhi + 1;

    v8f C = {};
#pragma unroll
    for (int s = 0; s < KSTEPS; ++s) {
      if (s >= slo && s <= shi) {                 // uniform per tile -> EXEC stays full
        v2f a;
        a.x = (s == 2 * n) ? A0.x : ((s == 2 * n + 1) ? A1.x : 0.0f);
        a.y = (s == 2 * n) ? A0.y : ((s == 2 * n + 1) ? A1.y : 0.0f);
        C = __builtin_amdgcn_wmma_f32_16x16x4_f32(
                /*neg_a=*/false, a, /*neg_b=*/false, B[s],
                /*c_mod=*/(short)0, C, /*reuse_a=*/false, /*reuse_b=*/false);
      }
    }

    // D layout: lane holds rows r0..r0+7 (+8 for lanes 16-31) at fixed h,
    // contiguous in memory -> two b128 stores; lane pair completes a 64B line.
    if (hok) {
      float* dst = out + (size_t)h * (size_t)NROWS + (size_t)(r0 + hi16 * 8);
      v4f lo = { C[0], C[1], C[2], C[3] };
      v4f hi = { C[4], C[5], C[6], C[7] };
      *(v4f*)(dst)     = lo;
      *(v4f*)(dst + 4) = hi;
    }
  }
}

extern "C" void kernel_launch(void* const* d_in, const int* in_sizes, int n_in,
                              void* d_out, int out_size, void* d_ws, size_t ws_size,
                              hipStream_t stream) {
  const float* x   = (const float*)d_in[0];   // (1, 4096, 7, 1) f32
  const float* ker = (const float*)d_in[1];   // (586, 8, 3) f32
  float* out = (float*)d_out;                 // 4096*12289 f32

  dim3 grid((HP + 15) / 16, 4, 1);            // 769 h-tiles x 4 row groups
  dim3 block(256, 1, 1);                      // 8 waves (wave32)
  conv_wmma_f32<<<grid, block, 0, stream>>>(x, ker, out);
}